// BlinnPhongShaderEnvMap_49486613184538
// MI455X (gfx1250) — compile-verified
//
#include <hip/hip_runtime.h>
#include <math.h>

#define HW_PIX (256 * 256)
#define NJ 1024
#define NLB 2

typedef __attribute__((ext_vector_type(2))) float v2f;
typedef __attribute__((ext_vector_type(8))) float v8f;

// ---------------------------------------------------------------------------
// Kernel 1: interpolate + normalize vertex normals per pixel.
// Writes pn padded to float4 into workspace (for WMMA A-operand loads) and
// writes the second reference output (pn broadcast over LB) directly.
// ---------------------------------------------------------------------------
__global__ __launch_bounds__(256) void pn_kernel(
    const float* __restrict__ verts_normals,  // (V,3)
    const int*   __restrict__ faces,          // (F,3)
    const int*   __restrict__ pix_to_face,    // (HW)
    const float* __restrict__ bary,           // (HW,3)
    float*       __restrict__ pn_ws,          // (HW,4) padded
    float*       __restrict__ out_pn)         // (LB,HW,3)
{
  const int p = blockIdx.x * 256 + threadIdx.x;  // exact grid: HW/256 blocks

  const int f  = pix_to_face[p];
  const int i0 = faces[f * 3 + 0];
  const int i1 = faces[f * 3 + 1];
  const int i2 = faces[f * 3 + 2];

  const float b0 = bary[p * 3 + 0];
  const float b1 = bary[p * 3 + 1];
  const float b2 = bary[p * 3 + 2];

  float x = b0 * verts_normals[i0 * 3 + 0] + b1 * verts_normals[i1 * 3 + 0] +
            b2 * verts_normals[i2 * 3 + 0];
  float y = b0 * verts_normals[i0 * 3 + 1] + b1 * verts_normals[i1 * 3 + 1] +
            b2 * verts_normals[i2 * 3 + 1];
  float z = b0 * verts_normals[i0 * 3 + 2] + b1 * verts_normals[i1 * 3 + 2] +
            b2 * verts_normals[i2 * 3 + 2];

  const float n   = sqrtf(x * x + y * y + z * z);
  const float inv = 1.0f / fmaxf(n, 1e-6f);
  x *= inv; y *= inv; z *= inv;

  pn_ws[p * 4 + 0] = x;
  pn_ws[p * 4 + 1] = y;
  pn_ws[p * 4 + 2] = z;
  pn_ws[p * 4 + 3] = 0.0f;  // K=3 padded to 4 for V_WMMA_F32_16X16X4_F32

  // Second output: pn is identical across the light-batch dimension.
  out_pn[p * 3 + 0] = x;
  out_pn[p * 3 + 1] = y;
  out_pn[p * 3 + 2] = z;
  out_pn[(size_t)HW_PIX * 3 + p * 3 + 0] = x;
  out_pn[(size_t)HW_PIX * 3 + p * 3 + 1] = y;
  out_pn[(size_t)HW_PIX * 3 + p * 3 + 2] = z;
}

// ---------------------------------------------------------------------------
// Kernel 2: fused attention-style shading.
//   diffuse(16pix x 16j) = WMMA_F32_16x16x4( pn_tile, light_dirs_tile )
//   clamp -> per-lane AV accumulation against LDS light colors
//   butterfly reduce over the 16 column lanes -> *texels -> out
// Block = 256 threads = 8 wave32s = 128 pixels; grid = (HW/128, LB).
// ---------------------------------------------------------------------------
__global__ __launch_bounds__(256) void shade_kernel(
    const float* __restrict__ pn_ws,       // (HW,4)
    const float* __restrict__ light_dirs,  // (LB,J,3)
    const float* __restrict__ env_map,     // (LB,J,3)
    const float* __restrict__ texels,      // (LB,HW,3)
    const float* __restrict__ kd,          // (1)
    float*       __restrict__ out_colors)  // (LB,HW,3)
{
  __shared__ float s_ld[NJ * 4];  // light dirs, padded float4   (16 KB)
  __shared__ float s_lc[NJ * 4];  // env_map * clamp(kd), padded (16 KB)

  const int b = blockIdx.y;
  const float kdc = fminf(fmaxf(kd[0], 0.0f), 1.0f);

  for (int j = threadIdx.x; j < NJ; j += 256) {
    const float* L = light_dirs + ((size_t)b * NJ + j) * 3;
    s_ld[j * 4 + 0] = L[0];
    s_ld[j * 4 + 1] = L[1];
    s_ld[j * 4 + 2] = L[2];
    s_ld[j * 4 + 3] = 0.0f;
    const float* E = env_map + ((size_t)b * NJ + j) * 3;
    s_lc[j * 4 + 0] = E[0] * kdc;
    s_lc[j * 4 + 1] = E[1] * kdc;
    s_lc[j * 4 + 2] = E[2] * kdc;
    s_lc[j * 4 + 3] = 0.0f;
  }
  __syncthreads();

  const int wave = threadIdx.x >> 5;
  const int lane = threadIdx.x & 31;
  const int lo   = lane & 15;   // column / row-within-tile index
  const int hi   = lane >> 4;   // half-wave select (K pair / M+8)

  const int base  = (blockIdx.x * 8 + wave) * 16;  // first pixel of tile
  const int m_pix = base + lo;                     // A-operand row pixel

  // A operand (16x4 f32): lanes 0-15 hold (K0,K1)=(x,y), lanes 16-31 (K2,K3)=(z,0)
  v2f a;
  a.x = pn_ws[m_pix * 4 + hi * 2 + 0];
  a.y = pn_ws[m_pix * 4 + hi * 2 + 1];

  float accx[8], accy[8], accz[8];
#pragma unroll
  for (int r = 0; r < 8; ++r) { accx[r] = 0.0f; accy[r] = 0.0f; accz[r] = 0.0f; }

#pragma unroll 4
  for (int jt = 0; jt < NJ / 16; ++jt) {
    const int j = jt * 16 + lo;

    // B operand (4x16 f32): lane's column j; lanes 0-15 (K0,K1), 16-31 (K2,K3)
    v2f bv;
    bv.x = s_ld[j * 4 + hi * 2 + 0];
    bv.y = s_ld[j * 4 + hi * 2 + 1];

    v8f c = {};
    v8f d = __builtin_amdgcn_wmma_f32_16x16x4_f32(
        /*neg_a=*/false, a, /*neg_b=*/false, bv,
        /*c_mod=*/(short)0, c, /*reuse_a=*/false, /*reuse_b=*/false);

    // Light color for this lane's column j (same for all 8 rows).
    const float cx = s_lc[j * 4 + 0];
    const float cy = s_lc[j * 4 + 1];
    const float cz = s_lc[j * 4 + 2];

    // D layout: VGPR r <-> row m = r + 8*hi, lane lo <-> column j.
#pragma unroll
    for (int r = 0; r < 8; ++r) {
      const float dv = fminf(fmaxf(d[r], 0.0f), 1.0f);  // clamp diffuse
      accx[r] += dv * cx;
      accy[r] += dv * cy;
      accz[r] += dv * cz;
    }
  }

  // Reduce over the 16 column lanes (xor <16 keeps hi-group invariant).
#pragma unroll
  for (int off = 1; off < 16; off <<= 1) {
#pragma unroll
    for (int r = 0; r < 8; ++r) {
      accx[r] += __shfl_xor(accx[r], off, 32);
      accy[r] += __shfl_xor(accy[r], off, 32);
      accz[r] += __shfl_xor(accz[r], off, 32);
    }
  }

  if (lo == 0) {
#pragma unroll
    for (int r = 0; r < 8; ++r) {
      const int pix = base + r + 8 * hi;  // rows 0-7 (hi=0), 8-15 (hi=1)
      const size_t o = ((size_t)b * HW_PIX + pix) * 3;
      out_colors[o + 0] = accx[r] * texels[o + 0];
      out_colors[o + 1] = accy[r] * texels[o + 1];
      out_colors[o + 2] = accz[r] * texels[o + 2];
    }
  }
}

// ---------------------------------------------------------------------------
extern "C" void kernel_launch(void* const* d_in, const int* in_sizes, int n_in,
                              void* d_out, int out_size, void* d_ws, size_t ws_size,
                              hipStream_t stream) {
  const float* verts_normals = (const float*)d_in[0];
  const int*   faces         = (const int*)d_in[1];
  const int*   pix_to_face   = (const int*)d_in[2];
  const float* bary          = (const float*)d_in[3];
  const float* light_dirs    = (const float*)d_in[4];
  const float* env_map       = (const float*)d_in[5];
  const float* texels        = (const float*)d_in[6];
  const float* kd            = (const float*)d_in[7];

  float* out_colors = (float*)d_out;                                  // (LB,H,W,3)
  float* out_pn     = (float*)d_out + (size_t)NLB * HW_PIX * 3;       // (LB,H,W,3)
  float* pn_ws      = (float*)d_ws;                                   // (HW,4) = 1 MB

  pn_kernel<<<HW_PIX / 256, 256, 0, stream>>>(
      verts_normals, faces, pix_to_face, bary, pn_ws, out_pn);

  shade_kernel<<<dim3(HW_PIX / 128, NLB), 256, 0, stream>>>(
      pn_ws, light_dirs, env_map, texels, kd, out_colors);
}